// RobustGeneGNN_85023172592042
// MI455X (gfx1250) — compile-verified
//
#include <hip/hip_runtime.h>
#include <hip/hip_bf16.h>

#define NN 50000
#define EE 800000

typedef __attribute__((ext_vector_type(2))) float v2f;
typedef __attribute__((ext_vector_type(8))) float v8f;

// ---------------------------------------------------------------------------
// GEMM: Hout[M,Fout] = X[M,K] @ W[K,Fout], fp32 WMMA 16x16x4.
// One wave computes a 16x64 tile (4 accumulators reuse one A fragment).
// grid = (M/16, Fout/64), blockDim = 32.
// A-frag (ISA 7.12.2, 32-bit A 16x4): lanes 0-15: M=lane, vgpr{0,1}=K{k,k+1};
// lanes 16-31: M=lane-16, vgpr{0,1}=K{k+2,k+3}.
// B-frag (4x16): vgpr v, laneHalf h -> K row = k + v + 2*h, N col = lane&15.
// C/D: vgpr v, laneHalf h -> row = v + 8*h, col = lane&15.
// ---------------------------------------------------------------------------
__global__ void gat_gemm_wmma(const float* __restrict__ X, const float* __restrict__ W,
                              float* __restrict__ Hout, int M, int K, int Fout) {
  const int lane = threadIdx.x & 31;
  const int half = lane >> 4;   // 0 or 1
  const int l16  = lane & 15;
  const int m0 = blockIdx.x * 16;
  const int n0 = blockIdx.y * 64;

  const float* __restrict__ xrow  = X + (size_t)(m0 + l16) * K;
  const float* __restrict__ wbase = W + n0 + l16;

  v8f acc0 = {}, acc1 = {}, acc2 = {}, acc3 = {};
  for (int k = 0; k < K; k += 4) {
    const int ka = k + half * 2;                       // even -> 8B aligned
    v2f a = *(const v2f*)(xrow + ka);                  // X[row, ka..ka+1]
    const float* __restrict__ w0 = wbase + (size_t)(ka + 0) * Fout;
    const float* __restrict__ w1 = wbase + (size_t)(ka + 1) * Fout;
    v2f b0, b1, b2, b3;
    b0.x = w0[0];  b0.y = w1[0];
    b1.x = w0[16]; b1.y = w1[16];
    b2.x = w0[32]; b2.y = w1[32];
    b3.x = w0[48]; b3.y = w1[48];
    acc0 = __builtin_amdgcn_wmma_f32_16x16x4_f32(false, a, false, b0, (short)0, acc0, false, false);
    acc1 = __builtin_amdgcn_wmma_f32_16x16x4_f32(false, a, false, b1, (short)0, acc1, false, false);
    acc2 = __builtin_amdgcn_wmma_f32_16x16x4_f32(false, a, false, b2, (short)0, acc2, false, false);
    acc3 = __builtin_amdgcn_wmma_f32_16x16x4_f32(false, a, false, b3, (short)0, acc3, false, false);
  }

  float* __restrict__ crow = Hout + (size_t)(m0 + half * 8) * Fout + n0 + l16;
#pragma unroll
  for (int v = 0; v < 8; ++v) {
    crow[(size_t)v * Fout + 0]  = acc0[v];
    crow[(size_t)v * Fout + 16] = acc1[v];
    crow[(size_t)v * Fout + 32] = acc2[v];
    crow[(size_t)v * Fout + 48] = acc3[v];
  }
}

// ---------------------------------------------------------------------------
// als[n,h] = sum_c H[n,h,c]*a_src[h,c]; ald likewise.
// ---------------------------------------------------------------------------
__global__ void gat_attn_coef(const float* __restrict__ Hf,
                              const float* __restrict__ a_src,
                              const float* __restrict__ a_dst,
                              float* __restrict__ als, float* __restrict__ ald,
                              int n_nodes, int heads, int C) {
  int idx = blockIdx.x * blockDim.x + threadIdx.x;
  if (idx >= n_nodes * heads) return;
  const int h = idx % heads;
  const float* __restrict__ hp = Hf + (size_t)idx * C;
  const float* __restrict__ as = a_src + h * C;
  const float* __restrict__ ad = a_dst + h * C;
  float ss = 0.f, sd = 0.f;
  for (int c = 0; c < C; ++c) {
    float v = hp[c];
    ss += v * as[c];
    sd += v * ad[c];
  }
  als[idx] = ss;
  ald[idx] = sd;
}

// zero agg[0..aggN), m = -inf, den = 0 over [0..mN)
__global__ void gat_init(float* __restrict__ agg, float* __restrict__ m,
                         float* __restrict__ den, int aggN, int mN) {
  int idx = blockIdx.x * blockDim.x + threadIdx.x;
  if (idx < aggN) agg[idx] = 0.0f;
  if (idx < mN) {
    m[idx]   = -__builtin_inff();
    den[idx] = 0.0f;
  }
}

__device__ inline float lrelu02(float v) { return v > 0.0f ? v : 0.2f * v; }

// float atomic max via signed/unsigned ordering trick
__device__ inline void atomicMaxF(float* addr, float val) {
  if (val >= 0.0f)
    atomicMax((int*)addr, __float_as_int(val));
  else
    atomicMin((unsigned int*)addr, __float_as_uint(val));
}

// pass A: segment max of leaky-relu(e) over dst
__global__ void gat_edge_max(const int* __restrict__ src, const int* __restrict__ dst,
                             const float* __restrict__ als, const float* __restrict__ ald,
                             float* __restrict__ m, int n_edges, int heads) {
  int idx = blockIdx.x * blockDim.x + threadIdx.x;
  if (idx >= n_edges * heads) return;
  const int e = idx / heads, h = idx % heads;
  const int s = src[e], d = dst[e];
  float v = lrelu02(als[s * heads + h] + ald[d * heads + h]);
  atomicMaxF(&m[d * heads + h], v);
}

// pass B: denom = segment_sum(exp(e - m[dst]))
__global__ void gat_edge_expsum(const int* __restrict__ src, const int* __restrict__ dst,
                                const float* __restrict__ als, const float* __restrict__ ald,
                                const float* __restrict__ m, float* __restrict__ den,
                                int n_edges, int heads) {
  int idx = blockIdx.x * blockDim.x + threadIdx.x;
  if (idx >= n_edges * heads) return;
  const int e = idx / heads, h = idx % heads;
  const int s = src[e], d = dst[e];
  float v = lrelu02(als[s * heads + h] + ald[d * heads + h]);
  atomicAdd(&den[d * heads + h], __expf(v - m[d * heads + h]));
}

// pass C: agg[dst] += h[src] * alpha   (one wave per edge)
__global__ void gat_edge_scatter(const int* __restrict__ src, const int* __restrict__ dst,
                                 const float* __restrict__ als, const float* __restrict__ ald,
                                 const float* __restrict__ m, const float* __restrict__ den,
                                 const float* __restrict__ Hf, float* __restrict__ agg,
                                 int n_edges, int heads, int C) {
  const int wave = (blockIdx.x * blockDim.x + threadIdx.x) >> 5;
  const int lane = threadIdx.x & 31;
  if (wave >= n_edges) return;
  const int s = src[wave], d = dst[wave];
  const float* __restrict__ hs = Hf + (size_t)s * heads * C;
  float* __restrict__ ag = agg + (size_t)d * heads * C;
  for (int h = 0; h < heads; ++h) {
    float v = lrelu02(als[s * heads + h] + ald[d * heads + h]);
    float alpha = __expf(v - m[d * heads + h]) / (den[d * heads + h] + 1e-16f);
    for (int c = lane; c < C; c += 32)
      atomicAdd(&ag[h * C + c], hs[h * C + c] * alpha);
  }
}

// out = elu(agg + b)   (in-place safe: out may alias agg)
__global__ void gat_bias_elu(const float* __restrict__ agg, const float* __restrict__ b,
                             float* __restrict__ out, int n_nodes, int F) {
  int idx = blockIdx.x * blockDim.x + threadIdx.x;
  if (idx >= n_nodes * F) return;
  float v = agg[idx] + b[idx % F];
  out[idx] = v > 0.0f ? v : (__expf(v) - 1.0f);
}

// ---------------------------------------------------------------------------
static void run_gat_layer(const float* feat_in, const float* Wm, const float* as,
                          const float* ad, const float* bias,
                          float* Hbuf, float* agg, float* out,
                          float* ALS, float* ALD, float* Mb, float* DEN,
                          const int* src, const int* dst,
                          int K, int heads, int C, hipStream_t stream) {
  const int Fout = heads * C;
  dim3 gg(NN / 16, Fout / 64);
  gat_gemm_wmma<<<gg, 32, 0, stream>>>(feat_in, Wm, Hbuf, NN, K, Fout);

  const int nh = NN * heads;
  gat_attn_coef<<<(nh + 255) / 256, 256, 0, stream>>>(Hbuf, as, ad, ALS, ALD, NN, heads, C);

  const int aggN = NN * Fout;   // agg may alias feat_in: GEMM already consumed it
  gat_init<<<(aggN + 255) / 256, 256, 0, stream>>>(agg, Mb, DEN, aggN, nh);

  const int eh = EE * heads;
  gat_edge_max<<<(eh + 255) / 256, 256, 0, stream>>>(src, dst, ALS, ALD, Mb, EE, heads);
  gat_edge_expsum<<<(eh + 255) / 256, 256, 0, stream>>>(src, dst, ALS, ALD, Mb, DEN, EE, heads);

  const long long nthr = (long long)EE * 32;
  gat_edge_scatter<<<(int)((nthr + 255) / 256), 256, 0, stream>>>(
      src, dst, ALS, ALD, Mb, DEN, Hbuf, agg, EE, heads, C);

  gat_bias_elu<<<(aggN + 255) / 256, 256, 0, stream>>>(agg, bias, out, NN, Fout);
}

extern "C" void kernel_launch(void* const* d_in, const int* in_sizes, int n_in,
                              void* d_out, int out_size, void* d_ws, size_t ws_size,
                              hipStream_t stream) {
  const float* x   = (const float*)d_in[0];
  const int*   ei  = (const int*)d_in[1];
  const int*   src = ei;
  const int*   dst = ei + EE;
  const float* W1  = (const float*)d_in[2];
  const float* as1 = (const float*)d_in[3];
  const float* ad1 = (const float*)d_in[4];
  const float* b1  = (const float*)d_in[5];
  const float* W2  = (const float*)d_in[6];
  const float* as2 = (const float*)d_in[7];
  const float* ad2 = (const float*)d_in[8];
  const float* b2  = (const float*)d_in[9];
  const float* W3  = (const float*)d_in[10];
  const float* as3 = (const float*)d_in[11];
  const float* ad3 = (const float*)d_in[12];
  const float* b3  = (const float*)d_in[13];

  float* ws   = (float*)d_ws;
  float* FEAT = ws;                               // N*256 (agg aliases this)
  float* Hbuf = ws + (size_t)NN * 256;            // N*256
  float* ALS  = ws + (size_t)2 * NN * 256;        // N*4
  float* ALD  = ALS + (size_t)NN * 4;
  float* Mb   = ALD + (size_t)NN * 4;
  float* DEN  = Mb  + (size_t)NN * 4;

  // Layer 1: x[N,256] -> FEAT[N,256]  (4 heads x 64, concat)
  run_gat_layer(x,    W1, as1, ad1, b1, Hbuf, FEAT, FEAT, ALS, ALD, Mb, DEN,
                src, dst, 256, 4, 64, stream);
  // Layer 2: FEAT -> FEAT  (GEMM reads FEAT before init zeroes it as agg)
  run_gat_layer(FEAT, W2, as2, ad2, b2, Hbuf, FEAT, FEAT, ALS, ALD, Mb, DEN,
                src, dst, 256, 4, 64, stream);
  // Layer 3: FEAT -> d_out  (1 head x 64, mean over 1 head == identity)
  run_gat_layer(FEAT, W3, as3, ad3, b3, Hbuf, (float*)d_out, (float*)d_out,
                ALS, ALD, Mb, DEN, src, dst, 256, 1, 64, stream);
}